// FourierLayer_21492016349905
// MI455X (gfx1250) — compile-verified
//
#include <hip/hip_runtime.h>
#include <hip/hip_bf16.h>
#include <math.h>

// ---------------------------------------------------------------------------
// FourierLayer for MI455X (gfx1250, wave32).
//   x: (8, 1024, 256) f32  ->  out: (8, 1280, 256) f32
// Stage 1: DFT-as-matmul with V_WMMA_F32_16X16X4_F32 (f32 precision needed
//          for a 1024-term reduction feeding a top-k).
// Stage 2: per-series top-32 via register-resident wave32 selection.
// Stage 3: table-based cosine synthesis (exact integer phase mod 1024).
// ---------------------------------------------------------------------------

typedef float v2f __attribute__((ext_vector_type(2)));
typedef float v8f __attribute__((ext_vector_type(8)));

#define T_IN   1024
#define D_MOD  256
#define N_B    8
#define N_F    512          // compute freqs 0..511 (DC ignored, Nyquist excluded)
#define PRED   256
#define T_OUT  (T_IN + PRED)
#define KSEL   32
#define TWO_PI_OVER_T 0.006135923151542565f  // 2*pi/1024

// ---------------------------------------------------------------------------
// Kernel 1: Re[b,f,d] =  sum_t cos(2*pi*f*t/1024) * x[b,t,d]
//           Im[b,f,d] = -sum_t sin(2*pi*f*t/1024) * x[b,t,d]   (rfft sign)
// One wave per 16(f) x 16(d) tile; K loop over t in steps of 4 with f32 WMMA.
// ---------------------------------------------------------------------------
__global__ __launch_bounds__(32)
void fourier_dft_wmma(const float* __restrict__ x,
                      float* __restrict__ wsre,
                      float* __restrict__ wsim) {
  __shared__ float tc[T_IN];
  __shared__ float tsn[T_IN];
  const int lane = threadIdx.x;

  // Twiddle tables: cos/sin(2*pi*u/1024), u = 0..1023.
  for (int u = lane; u < T_IN; u += 32) {
    float ang = (float)u * TWO_PI_OVER_T;
    tc[u]  = cosf(ang);
    tsn[u] = sinf(ang);
  }
  __syncthreads();

  const int d0 = blockIdx.x << 4;         // d tile (16)
  const int f0 = blockIdx.y << 4;         // f tile (16)
  const int b  = blockIdx.z;

  const float* xb = x + (size_t)b * T_IN * D_MOD;

  const int m     = lane & 15;            // row within A / col within B,C,D
  const int khalf = (lane >> 4) << 1;     // 0 (lanes 0-15) or 2 (lanes 16-31)
  const int frow  = f0 + m;               // this lane's A-matrix row (freq)

  // B operand base: element (t = k + khalf [+1], d = d0 + m)
  const float* px = xb + (size_t)khalf * D_MOD + d0 + m;

  v8f accRe = {};
  v8f accIm = {};

  for (int k = 0; k < T_IN; k += 4) {
    // A layout (16x4 f32): VGPR0 -> K = khalf, VGPR1 -> K = khalf+1
    const int kk = k + khalf;
    const int w0 = (frow * kk)       & (T_IN - 1);
    const int w1 = (frow * (kk + 1)) & (T_IN - 1);

    v2f aC, aS, bV;
    aC[0] = tc[w0];   aC[1] = tc[w1];     // cos rows  -> Re
    aS[0] = -tsn[w0]; aS[1] = -tsn[w1];   // -sin rows -> Im (rfft e^{-i...})

    // B layout (4x16 f32), mirrored: VGPR0 -> K = khalf, VGPR1 -> K = khalf+1
    bV[0] = px[k * D_MOD];
    bV[1] = px[k * D_MOD + D_MOD];

    accRe = __builtin_amdgcn_wmma_f32_16x16x4_f32(
        false, aC, false, bV, (short)0, accRe, false, false);
    accIm = __builtin_amdgcn_wmma_f32_16x16x4_f32(
        false, aS, false, bV, (short)0, accIm, false, false);
  }

  // C/D layout: VGPR j -> M = j (lanes 0-15) or M = j+8 (lanes 16-31), N = lane%16
  const int rbase = (lane >> 4) << 3;     // 0 or 8
  const int col   = d0 + m;
#pragma unroll
  for (int j = 0; j < 8; ++j) {
    const int row = f0 + rbase + j;
    const size_t o = ((size_t)b * N_F + row) * D_MOD + col;
    wsre[o] = accRe[j];
    wsim[o] = accIm[j];
  }
}

// ---------------------------------------------------------------------------
// Kernel 2: per-series (b,d) top-32 over |X[f]|^2, f in 1..511.
// One wave per series; 16 candidate magnitudes live in registers per lane.
// 32 rounds of wave-max (lowest index wins ties, matching jax top_k).
// Emits m (freq index), a = 2*Re/1024, c = -2*Im/1024.
// ---------------------------------------------------------------------------
__global__ __launch_bounds__(32)
void fourier_topk(const float* __restrict__ wsre,
                  const float* __restrict__ wsim,
                  int*   __restrict__ pm,
                  float* __restrict__ pa,
                  float* __restrict__ pb) {
  const int s = blockIdx.x;               // series 0..2047
  const int b = s >> 8;
  const int d = s & (D_MOD - 1);
  const int lane = threadIdx.x;

  const float* rb = wsre + (size_t)b * N_F * D_MOD + d;
  const float* ib = wsim + (size_t)b * N_F * D_MOD + d;

  float mag[16];
#pragma unroll
  for (int j = 0; j < 16; ++j) {
    const int f = 1 + j * 32 + lane;      // covers 1..512; 512 excluded below
    if (f <= 511) {
      const float re = rb[(size_t)f * D_MOD];
      const float im = ib[(size_t)f * D_MOD];
      mag[j] = re * re + im * im;
    } else {
      mag[j] = -1.0f;
    }
  }

  int myf = 1;
  for (int r = 0; r < KSEL; ++r) {
    // local best (lowest j on tie via strict >)
    float bv = -3.0f;
    int bj = 0;
#pragma unroll
    for (int j = 0; j < 16; ++j) {
      if (mag[j] > bv) { bv = mag[j]; bj = j; }
    }
    float v = bv;
    int gi = 1 + bj * 32 + lane;          // global freq index

    // wave32 reduce: max value, lowest index on tie
#pragma unroll
    for (int off = 16; off >= 1; off >>= 1) {
      const float ov = __shfl_xor(v, off, 32);
      const int   oi = __shfl_xor(gi, off, 32);
      if (ov > v || (ov == v && oi < gi)) { v = ov; gi = oi; }
    }

    if (lane == r) myf = gi;              // lane r owns result r

    // winner lane retires its entry
    const int wl = (gi - 1) & 31;
    const int wj = (gi - 1) >> 5;
    if (lane == wl) {
#pragma unroll
      for (int j = 0; j < 16; ++j)
        if (j == wj) mag[j] = -2.0f;
    }
  }

  const float re = rb[(size_t)myf * D_MOD];
  const float im = ib[(size_t)myf * D_MOD];
  // amp*cos(phi) = Re/T, amp*sin(phi) = Im/T; conj half doubles -> /512
  pm[s * KSEL + lane] = myf;
  pa[s * KSEL + lane] = re * (1.0f / 512.0f);
  pb[s * KSEL + lane] = -im * (1.0f / 512.0f);
}

// ---------------------------------------------------------------------------
// Kernel 3: out[b,t,d] = sum_k a_k*cos(2*pi*m_k*t/1024) + c_k*sin(...)
// Integer phase (m*t) & 1023 -> exact; LDS tables; coalesced stores over d.
// ---------------------------------------------------------------------------
#define TT_TILE 64
__global__ __launch_bounds__(256)
void fourier_synth(const int*   __restrict__ pm,
                   const float* __restrict__ pa,
                   const float* __restrict__ pb,
                   float* __restrict__ out) {
  __shared__ float tc[T_IN];
  __shared__ float tsn[T_IN];
  const int d = threadIdx.x;
  for (int u = d; u < T_IN; u += 256) {
    float ang = (float)u * TWO_PI_OVER_T;
    tc[u]  = cosf(ang);
    tsn[u] = sinf(ang);
  }
  __syncthreads();

  const int b  = blockIdx.y;
  const int t0 = blockIdx.x * TT_TILE;
  const int s  = (b << 8) | d;

  int   m[KSEL];
  float a[KSEL];
  float c[KSEL];
#pragma unroll
  for (int k = 0; k < KSEL; ++k) {
    m[k] = pm[s * KSEL + k];
    a[k] = pa[s * KSEL + k];
    c[k] = pb[s * KSEL + k];
  }

  for (int tl = 0; tl < TT_TILE; ++tl) {
    const int tt = t0 + tl;
    float acc = 0.0f;
#pragma unroll
    for (int k = 0; k < KSEL; ++k) {
      const int u = (m[k] * tt) & (T_IN - 1);
      acc += a[k] * tc[u] + c[k] * tsn[u];
    }
    out[((size_t)b * T_OUT + tt) * D_MOD + d] = acc;
  }
}

// ---------------------------------------------------------------------------
extern "C" void kernel_launch(void* const* d_in, const int* in_sizes, int n_in,
                              void* d_out, int out_size, void* d_ws, size_t ws_size,
                              hipStream_t stream) {
  (void)in_sizes; (void)n_in; (void)out_size; (void)ws_size;

  const float* x = (const float*)d_in[0];
  float* out = (float*)d_out;

  char* ws = (char*)d_ws;
  const size_t spec_bytes = (size_t)N_B * N_F * D_MOD * sizeof(float);   // 4 MB
  float* wsre = (float*)(ws);
  float* wsim = (float*)(ws + spec_bytes);
  char*  pbase = ws + 2 * spec_bytes;
  const size_t par_bytes = (size_t)N_B * D_MOD * KSEL * sizeof(float);   // 256 KB
  int*   pm = (int*)  (pbase);
  float* pa = (float*)(pbase + par_bytes);
  float* pb = (float*)(pbase + 2 * par_bytes);

  // Stage 1: 16 d-tiles x 32 f-tiles x 8 batches, one wave each.
  fourier_dft_wmma<<<dim3(D_MOD / 16, N_F / 16, N_B), 32, 0, stream>>>(x, wsre, wsim);

  // Stage 2: one wave per (b,d) series.
  fourier_topk<<<N_B * D_MOD, 32, 0, stream>>>(wsre, wsim, pm, pa, pb);

  // Stage 3: 20 time-tiles x 8 batches, 256 threads (= d) each.
  fourier_synth<<<dim3(T_OUT / TT_TILE, N_B), 256, 0, stream>>>(pm, pa, pb, out);
}